// GCN_ASAP_11218454577328
// MI455X (gfx1250) — compile-verified
//
#include <hip/hip_runtime.h>
#include <math.h>

typedef float v2f __attribute__((ext_vector_type(2)));
typedef float v8f __attribute__((ext_vector_type(8)));

#define NNODES 50000
#define NEDGES 800000
#define DIN    128
#define DHID   256
#define DOUTF  128
#define BN_EPS 1e-5f

static __device__ __forceinline__ void atomic_add_f32(float* p, float v) {
    __hip_atomic_fetch_add(p, v, __ATOMIC_RELAXED, __HIP_MEMORY_SCOPE_AGENT);
}

// ---------------- degree / norm ----------------
__global__ void k_degree(const int* __restrict__ ei, const float* __restrict__ ew,
                         float* __restrict__ deg, int E, int n) {
    int i = blockIdx.x * blockDim.x + threadIdx.x;
    if (i < E) {
        atomic_add_f32(&deg[ei[E + i]], ew[i]);       // col index
    } else if (i < E + n) {
        atomic_add_f32(&deg[i - E], 1.0f);            // self-loop weight 1
    }
}

__global__ void k_dinv(float* __restrict__ deg, int n) {
    int i = blockIdx.x * blockDim.x + threadIdx.x;
    if (i < n) {
        float d = deg[i];
        deg[i] = (d > 0.0f) ? rsqrtf(d) : 0.0f;
    }
}

__global__ void k_norm(const int* __restrict__ ei, const float* __restrict__ ew,
                       const float* __restrict__ dinv, float* __restrict__ norm,
                       int E, int n) {
    int e = blockIdx.x * blockDim.x + threadIdx.x;
    if (e < E) {
        norm[e] = dinv[ei[e]] * ew[e] * dinv[ei[E + e]];
    } else if (e < E + n) {
        float d = dinv[e - E];
        norm[e] = d * d;
    }
}

// ---------------- WMMA fp32 GEMM: C[M,Nc] = A[M,K] @ W[K,Nc] ----------------
// One wave -> (16*RB) x 64 tile. B fragments (8 scalar loads/step) are shared
// across RB row-blocks: RB=2 gives 8 WMMAs per 10 memory instructions.
template<int RB>
__global__ __launch_bounds__(128)
void k_gemm_wmma(const float* __restrict__ A, const float* __restrict__ W,
                 float* __restrict__ C, int M, int K, int Nc) {
    const int lane   = threadIdx.x & 31;
    const int wave   = threadIdx.x >> 5;
    const int tilesN = Nc >> 6;                     // 64-wide n groups
    const int tile   = blockIdx.x * 4 + wave;
    const int tm = tile / tilesN;
    const int tn = tile % tilesN;
    const int rowBase = tm * (16 * RB);
    if (rowBase >= M) return;

    const int hi = lane >> 4;                       // 0: K pair (0,1); 1: K pair (2,3)
    const int l  = lane & 15;
    const int row0 = rowBase + l;                   // A row, block 0
    const int col0 = tn * 64 + l;                   // first B/C column for this lane

    v8f acc[RB][4];
#pragma unroll
    for (int rb = 0; rb < RB; ++rb)
#pragma unroll
        for (int j = 0; j < 4; ++j) acc[rb][j] = (v8f){};

#pragma unroll 4
    for (int k0 = 0; k0 < K; k0 += 4) {
        const int ka = k0 + 2 * hi;
        // A 16x4 fragments: VGPR0=K=ka, VGPR1=K=ka+1
        v2f a[RB];
#pragma unroll
        for (int rb = 0; rb < RB; ++rb)
            a[rb] = *(const v2f*)(A + (size_t)(row0 + 16 * rb) * K + ka);

        // B 4x16 fragments, same K split across lane halves
        const float* Wr0 = W + (size_t)ka * Nc;
        const float* Wr1 = Wr0 + Nc;
        v2f b[4];
#pragma unroll
        for (int j = 0; j < 4; ++j) {
            b[j].x = Wr0[col0 + 16 * j];
            b[j].y = Wr1[col0 + 16 * j];
        }

#pragma unroll
        for (int rb = 0; rb < RB; ++rb)
#pragma unroll
            for (int j = 0; j < 4; ++j)
                acc[rb][j] = __builtin_amdgcn_wmma_f32_16x16x4_f32(
                    false, a[rb], false, b[j], (short)0, acc[rb][j], false, false);
    }

    // C/D layout: VGPR r, lanes 0-15 -> M=r; lanes 16-31 -> M=r+8
#pragma unroll
    for (int rb = 0; rb < RB; ++rb) {
#pragma unroll
        for (int r = 0; r < 8; ++r) {
            const int orow = rowBase + 16 * rb + r + 8 * hi;
            float* Crow = C + (size_t)orow * Nc + tn * 64 + l;
            Crow[0]  = acc[rb][0][r];
            Crow[16] = acc[rb][1][r];
            Crow[32] = acc[rb][2][r];
            Crow[48] = acc[rb][3][r];
        }
    }
}

// host-side: main body (32-row tiles over 49984 rows) + 16-row tail
static void launch_gemm(const float* A, const float* W, float* C,
                        int K, int Nc, hipStream_t stream) {
    const int M_MAIN = 49984;                       // 1562 * 32
    const int tilesN = Nc >> 6;
    {
        const int tiles = (M_MAIN / 32) * tilesN;
        k_gemm_wmma<2><<<(tiles + 3) / 4, 128, 0, stream>>>(A, W, C, M_MAIN, K, Nc);
    }
    {   // tail: rows [49984, 50000)
        const int tiles = tilesN;
        k_gemm_wmma<1><<<(tiles + 3) / 4, 128, 0, stream>>>(
            A + (size_t)M_MAIN * K, W, C + (size_t)M_MAIN * Nc, 16, K, Nc);
    }
}

// ---------------- edge scatter: out[col] += t[row] * norm ----------------
// one float4 feature chunk per thread; shiftC = log2(d/4)
__global__ void k_scatter(const float* __restrict__ t, const int* __restrict__ ei,
                          const float* __restrict__ norm, float* __restrict__ out,
                          int E, int n, int d, int shiftC) {
    long long idx = (long long)blockIdx.x * blockDim.x + threadIdx.x;
    const int chunks = 1 << shiftC;
    long long total = (long long)(E + n) * chunks;
    if (idx >= total) return;
    const int e = (int)(idx >> shiftC);
    const int f = ((int)idx & (chunks - 1)) << 2;
    int r, c;
    if (e < E) { r = ei[e]; c = ei[E + e]; } else { r = c = e - E; }
    const float w = norm[e];
    const float4 v = *(const float4*)(t + (size_t)r * d + f);
    float* o = out + (size_t)c * d + f;
    atomic_add_f32(o + 0, v.x * w);
    atomic_add_f32(o + 1, v.y * w);
    atomic_add_f32(o + 2, v.z * w);
    atomic_add_f32(o + 3, v.w * w);
}

// ---------------- fused bias + BN(eval) + ReLU ----------------
__global__ void k_bias_bn_relu(float* __restrict__ h, const float* __restrict__ b,
                               const float* __restrict__ g, const float* __restrict__ be,
                               long long total, int dmask) {
    long long i = (long long)blockIdx.x * blockDim.x + threadIdx.x;
    if (i >= total) return;
    const int j = (int)i & dmask;
    const float scale = g[j] * rsqrtf(1.0f + BN_EPS);
    float v = (h[i] + b[j]) * scale + be[j];
    h[i] = v > 0.0f ? v : 0.0f;
}

// ---------------- final bias + log_softmax over 128 features ----------------
__global__ __launch_bounds__(256)
void k_logsoftmax(float* __restrict__ out, const float* __restrict__ bias, int n) {
    const int wave = threadIdx.x >> 5;
    const int lane = threadIdx.x & 31;
    const int node = blockIdx.x * 8 + wave;
    if (node >= n) return;
    float* row = out + (size_t)node * DOUTF;
    float4 v = *(float4*)(row + lane * 4);
    const float4 b = *(const float4*)(bias + lane * 4);
    v.x += b.x; v.y += b.y; v.z += b.z; v.w += b.w;
    float m = fmaxf(fmaxf(v.x, v.y), fmaxf(v.z, v.w));
#pragma unroll
    for (int off = 16; off >= 1; off >>= 1) m = fmaxf(m, __shfl_xor(m, off, 32));
    float s = expf(v.x - m) + expf(v.y - m) + expf(v.z - m) + expf(v.w - m);
#pragma unroll
    for (int off = 16; off >= 1; off >>= 1) s += __shfl_xor(s, off, 32);
    const float lse = m + logf(s);
    v.x -= lse; v.y -= lse; v.z -= lse; v.w -= lse;
    *(float4*)(row + lane * 4) = v;
}

extern "C" void kernel_launch(void* const* d_in, const int* in_sizes, int n_in,
                              void* d_out, int out_size, void* d_ws, size_t ws_size,
                              hipStream_t stream) {
    const float* x   = (const float*)d_in[0];
    const int*   ei  = (const int*)  d_in[1];   // [2,E] flat: rows then cols
    const float* ew  = (const float*)d_in[2];
    const float* W0  = (const float*)d_in[3];
    const float* b0  = (const float*)d_in[4];
    const float* W1  = (const float*)d_in[5];
    const float* b1  = (const float*)d_in[6];
    const float* Wl  = (const float*)d_in[7];
    const float* bl  = (const float*)d_in[8];
    const float* g0  = (const float*)d_in[9];
    const float* be0 = (const float*)d_in[10];
    const float* g1  = (const float*)d_in[11];
    const float* be1 = (const float*)d_in[12];
    float* out = (float*)d_out;

    float* dinv = (float*)d_ws;                               // N
    float* norm = dinv + NNODES;                              // E+N
    float* bufA = norm + (NEDGES + NNODES);                   // N*DHID
    float* bufB = bufA + (size_t)NNODES * DHID;               // N*DHID

    const int tEN = NEDGES + NNODES;

    // ---- normalization coefficients ----
    hipMemsetAsync(dinv, 0, NNODES * sizeof(float), stream);
    k_degree<<<(tEN + 255) / 256, 256, 0, stream>>>(ei, ew, dinv, NEDGES, NNODES);
    k_dinv<<<(NNODES + 255) / 256, 256, 0, stream>>>(dinv, NNODES);
    k_norm<<<(tEN + 255) / 256, 256, 0, stream>>>(ei, ew, dinv, norm, NEDGES, NNODES);

    // ---- layer 0: x @ W0 -> bufA ; scatter -> bufB ; bias+BN+ReLU ----
    launch_gemm(x, W0, bufA, DIN, DHID, stream);
    hipMemsetAsync(bufB, 0, (size_t)NNODES * DHID * sizeof(float), stream);
    {
        long long total = (long long)tEN * (DHID / 4);
        k_scatter<<<(int)((total + 255) / 256), 256, 0, stream>>>(bufA, ei, norm, bufB,
                                                                  NEDGES, NNODES, DHID, 6);
    }
    {
        long long total = (long long)NNODES * DHID;
        k_bias_bn_relu<<<(int)((total + 255) / 256), 256, 0, stream>>>(bufB, b0, g0, be0,
                                                                       total, DHID - 1);
    }

    // ---- layer 1: bufB @ W1 -> bufA ; scatter -> bufB ; bias+BN+ReLU ----
    launch_gemm(bufB, W1, bufA, DHID, DHID, stream);
    hipMemsetAsync(bufB, 0, (size_t)NNODES * DHID * sizeof(float), stream);
    {
        long long total = (long long)tEN * (DHID / 4);
        k_scatter<<<(int)((total + 255) / 256), 256, 0, stream>>>(bufA, ei, norm, bufB,
                                                                  NEDGES, NNODES, DHID, 6);
    }
    {
        long long total = (long long)NNODES * DHID;
        k_bias_bn_relu<<<(int)((total + 255) / 256), 256, 0, stream>>>(bufB, b1, g1, be1,
                                                                       total, DHID - 1);
    }

    // ---- layer 2: bufB @ Wl -> bufA ; scatter -> d_out ; bias + log_softmax ----
    launch_gemm(bufB, Wl, bufA, DHID, DOUTF, stream);
    hipMemsetAsync(out, 0, (size_t)NNODES * DOUTF * sizeof(float), stream);
    {
        long long total = (long long)tEN * (DOUTF / 4);
        k_scatter<<<(int)((total + 255) / 256), 256, 0, stream>>>(bufA, ei, norm, out,
                                                                  NEDGES, NNODES, DOUTF, 5);
    }
    k_logsoftmax<<<(NNODES + 7) / 8, 256, 0, stream>>>(out, bl, NNODES);
}